// GreedyCTCDecoder_2989297238246
// MI455X (gfx1250) — compile-verified
//
#include <hip/hip_runtime.h>
#include <stdint.h>

// Greedy CTC decode for emission [T=4096, VOCAB=32000] f32.
// Output (int32): out[0..T-1] = left-packed decoded labels padded with -1,
//                 out[T]      = count.
//
// Kernel 1: per-row argmax, streamed through LDS with CDNA5 async copies.
// Kernel 2: collapse repeats, drop blanks, stream-compact (single block).

#define VOCAB        32000
#define BLANK        0
#define BLOCK1       320                     // 10 wave32s
#define F4_PER_ROW   (VOCAB / 4)             // 8000 float4 per row
#define CHUNKS       5
#define F4_PER_CHUNK (F4_PER_ROW / CHUNKS)   // 1600 float4 = 25600 B
#define J_PER_CHUNK  (F4_PER_CHUNK / BLOCK1) // 5 float4 per thread per chunk

// Low 32 bits of a generic pointer derived from LDS == byte offset from the
// wave's LDS base (addrspacecast(3->0) is {aperture_hi, lds_offset}).
__device__ __forceinline__ uint32_t lds_byte_off(const void* p) {
  return (uint32_t)(uint64_t)(uintptr_t)p;
}

__global__ __launch_bounds__(BLOCK1) void ctc_argmax_kernel(
    const float* __restrict__ emission, int* __restrict__ amax) {
  __shared__ float4 buf[2][F4_PER_CHUNK];      // 51200 B double buffer
  __shared__ float  sval[BLOCK1 / 32];
  __shared__ int    sidx[BLOCK1 / 32];

  const uint32_t t   = threadIdx.x;
  const uint32_t row = blockIdx.x;
  const float*   rowBase = emission + (size_t)row * VOCAB;

  auto issue_chunk = [&](int c) {
    const int cb = c & 1;
#pragma unroll
    for (int j = 0; j < J_PER_CHUNK; ++j) {
      uint32_t f4   = (uint32_t)c * F4_PER_CHUNK + (uint32_t)j * BLOCK1 + t;
      uint32_t voff = f4 * 16u;                                  // global byte offset
      uint32_t loff = lds_byte_off(&buf[cb][j * BLOCK1 + t]);    // LDS byte offset
      asm volatile("global_load_async_to_lds_b128 %0, %1, %2"
                   :: "v"(loff), "v"(voff), "s"(rowBase)
                   : "memory");
    }
  };

  float bv = -__builtin_inff();
  int   bi = 0;

  issue_chunk(0);
#pragma unroll
  for (int c = 0; c < CHUNKS; ++c) {
    if (c + 1 < CHUNKS) {
      // Previous-previous chunk's LDS reads must be drained before the async
      // engine overwrites that buffer; its data was already consumed by VALU.
      asm volatile("s_wait_dscnt 0" ::: "memory");
      issue_chunk(c + 1);
      asm volatile("s_wait_asynccnt 5" ::: "memory");  // chunk c resident
    } else {
      asm volatile("s_wait_asynccnt 0" ::: "memory");
    }
    const int cb = c & 1;
#pragma unroll
    for (int j = 0; j < J_PER_CHUNK; ++j) {
      float4 w  = buf[cb][j * BLOCK1 + (int)t];
      int    fb = (c * F4_PER_CHUNK + j * BLOCK1 + (int)t) * 4;
      // ascending index + strict '>' keeps the FIRST max (jnp.argmax tie rule)
      if (w.x > bv) { bv = w.x; bi = fb + 0; }
      if (w.y > bv) { bv = w.y; bi = fb + 1; }
      if (w.z > bv) { bv = w.z; bi = fb + 2; }
      if (w.w > bv) { bv = w.w; bi = fb + 3; }
    }
  }

  // wave32 (lexicographic: larger value, then smaller index)
#pragma unroll
  for (int m = 16; m >= 1; m >>= 1) {
    float ov = __shfl_xor(bv, m, 32);
    int   oi = __shfl_xor(bi, m, 32);
    if (ov > bv || (ov == bv && oi < bi)) { bv = ov; bi = oi; }
  }
  const int lane = (int)(t & 31);
  const int wid  = (int)(t >> 5);
  if (lane == 0) { sval[wid] = bv; sidx[wid] = bi; }
  __syncthreads();
  if (t == 0) {
    float fv = sval[0]; int fi = sidx[0];
#pragma unroll
    for (int w = 1; w < BLOCK1 / 32; ++w)
      if (sval[w] > fv || (sval[w] == fv && sidx[w] < fi)) { fv = sval[w]; fi = sidx[w]; }
    amax[row] = fi;
  }
}

__global__ __launch_bounds__(1024) void ctc_compact_kernel(
    const int* __restrict__ amax, int* __restrict__ out, int T) {
  const int tid  = (int)threadIdx.x;
  const int lane = tid & 31;
  const int wid  = tid >> 5;
  const int PT   = T >> 10;                 // elements per thread (4 for T=4096)

  int vals[8];
  int keep[8];
  int base = tid * PT;
  int prev = (base > 0) ? amax[base - 1] : -1;   // -1 never matches a label
  int cnt  = 0;
  for (int k = 0; k < PT; ++k) {
    int  v  = amax[base + k];
    int  kp = (v != prev && v != BLANK) ? 1 : 0;
    vals[k] = v; keep[k] = kp; cnt += kp;
    prev = v;
  }

  // inclusive wave scan of per-thread counts
  int inc = cnt;
#pragma unroll
  for (int off = 1; off < 32; off <<= 1) {
    int n = __shfl_up(inc, off, 32);
    if (lane >= off) inc += n;
  }

  __shared__ int wsum[32];
  __shared__ int woff[32];
  __shared__ int s_total;
  if (lane == 31) wsum[wid] = inc;
  __syncthreads();
  if (wid == 0) {
    int w  = wsum[lane];
    int wi = w;
#pragma unroll
    for (int off = 1; off < 32; off <<= 1) {
      int n = __shfl_up(wi, off, 32);
      if (lane >= off) wi += n;
    }
    woff[lane] = wi - w;                    // exclusive warp offset
    if (lane == 31) { s_total = wi; out[T] = wi; }
  }
  __syncthreads();

  const int total = s_total;
  int pos = woff[wid] + (inc - cnt);        // exclusive start for this thread

  // Disjoint coverage: scatter fills [0,total), pad fills [total,T).
  for (int k = 0; k < PT; ++k) {
    int j = base + k;
    if (j >= total) out[j] = -1;
  }
  for (int k = 0; k < PT; ++k) {
    if (keep[k]) { out[pos] = vals[k]; ++pos; }
  }
}

extern "C" void kernel_launch(void* const* d_in, const int* in_sizes, int n_in,
                              void* d_out, int out_size, void* d_ws, size_t ws_size,
                              hipStream_t stream) {
  (void)n_in; (void)out_size; (void)ws_size;
  const float* emission = (const float*)d_in[0];
  const int T = in_sizes[0] / VOCAB;        // 4096
  int* amax = (int*)d_ws;                   // T int32 scratch
  int* out  = (int*)d_out;                  // T labels + 1 count (int32)

  ctc_argmax_kernel<<<T, BLOCK1, 0, stream>>>(emission, amax);
  ctc_compact_kernel<<<1, 1024, 0, stream>>>(amax, out, T);
}